// RepulsionLoss_16020228014463
// MI455X (gfx1250) — compile-verified
//
#include <hip/hip_runtime.h>
#include <math.h>
#include <stdint.h>

// ---------------------------------------------------------------------------
// RepulsionLoss for N=4096 boxes, f32, on gfx1250 (MI455X).
// Pairwise phase: wave-per-16-row-tile; V_WMMA_F32_16X16X4_F32 computes the
// target Gram tile used as a same-target prefilter (bitwise compare confirms,
// under a single per-tile min-distance branch).
// ---------------------------------------------------------------------------

typedef float v2f __attribute__((ext_vector_type(2)));
typedef float v8f __attribute__((ext_vector_type(8)));

#define N_BOX        4096
#define ITILES       (N_BOX / 16)          // 256 i-tiles
#define WAVES_PER_BLK 8
#define JCHUNKS      8
#define JT_PER_CHUNK (ITILES / JCHUNKS)    // 32 j-tiles per chunk

__device__ inline unsigned long long shflxor_u64(unsigned long long x, int m) {
    unsigned int lo = (unsigned int)x, hi = (unsigned int)(x >> 32);
    lo = __shfl_xor(lo, m, 32);
    hi = __shfl_xor(hi, m, 32);
    return ((unsigned long long)hi << 32) | lo;
}

// Bitwise box equality: xor/or-combine lowers to v_xor + v_or3/v_bitop3.
// (inputs are uniform-random positive floats: no -0.0 / NaN, so bit
//  equality is float equality here)
__device__ inline bool box_bits_eq(float4 a, float4 b) {
    unsigned int d = (__float_as_uint(a.x) ^ __float_as_uint(b.x)) |
                     (__float_as_uint(a.y) ^ __float_as_uint(b.y)) |
                     (__float_as_uint(a.z) ^ __float_as_uint(b.z)) |
                     (__float_as_uint(a.w) ^ __float_as_uint(b.w));
    return d == 0u;
}

// ---------------- init workspace -------------------------------------------
__global__ void init_ws_kernel(unsigned long long* rowmax, unsigned int* hdr) {
    int t = blockIdx.x * blockDim.x + threadIdx.x;
    if (t < N_BOX) rowmax[t] = 0ull;
    if (t < 8)     hdr[t] = 0u;   // [0]=sl1 sum f32, [1]=rb sum f32, [2]=rb cnt,
                                  // [3]=rg sum f32, [4]=rg cnt
}

// ---------------- smooth-L1 (mean over N*4 elems, beta=1) ------------------
__global__ void sl1_kernel(const float* __restrict__ pred,
                           const float* __restrict__ target,
                           float* __restrict__ sl1_sum) {
    int t = blockIdx.x * blockDim.x + threadIdx.x;   // 4096 threads
    float s = 0.f;
    for (int e = t; e < N_BOX * 4; e += 4096) {
        float d = fabsf(pred[e] - target[e]);
        s += (d < 1.0f) ? 0.5f * d * d : d - 0.5f;
    }
    for (int m = 16; m >= 1; m >>= 1) s += __shfl_xor(s, m, 32);
    if ((threadIdx.x & 31) == 0) atomicAdd(sl1_sum, s);
}

// ---------------- pairwise IoU / RepGT-argmax / RepBox ---------------------
__global__ void __launch_bounds__(256)
pairwise_kernel(const float* __restrict__ pred,
                const float* __restrict__ target,
                unsigned long long* __restrict__ rowmax,
                float* __restrict__ rb_sum,
                unsigned int* __restrict__ rb_cnt) {
    const int lane  = threadIdx.x & 31;
    const int wave  = threadIdx.x >> 5;
    const int itile = blockIdx.x * WAVES_PER_BLK + wave;
    const int i0    = itile * 16;
    const int j0base = (int)blockIdx.y * JT_PER_CHUNK * 16;

    const float4* P4 = (const float4*)pred;
    const float4* T4 = (const float4*)target;

    const int r  = lane & 15;   // column / row-in-tile index for frags
    const int kh = lane >> 4;   // K-half (A/B frag), row-half (C/D frag)

    // A fragment: target rows of the i-tile (16x4, K=4).
    // Layout: lanes 0-15 carry K={0,1}, lanes 16-31 carry K={2,3}.
    float4 ti = T4[i0 + r];
    v2f afrag;
    afrag.x = kh ? ti.z : ti.x;
    afrag.y = kh ? ti.w : ti.y;
    float q_lane = ti.x * ti.x + ti.y * ti.y + ti.z * ti.z + ti.w * ti.w;

    // Per-lane loop-invariant state for the 8 rows m = v + 8*kh:
    //   pred box + area (+1 convention), target box (exact-compare confirm),
    //   target squared norm (shuffled from the A-frag lanes).
    float4 p[8], ta[8];
    float  parea[8], qi[8];
#pragma unroll
    for (int v = 0; v < 8; ++v) {
        p[v]     = P4[i0 + v + 8 * kh];
        ta[v]    = T4[i0 + v + 8 * kh];
        parea[v] = (p[v].z - p[v].x + 1.f) * (p[v].w - p[v].y + 1.f);
        qi[v]    = __shfl(q_lane, v + 8 * kh, 32);
    }

    unsigned long long rmax[8];
#pragma unroll
    for (int v = 0; v < 8; ++v) rmax[v] = 0ull;
    float rbs = 0.f;
    unsigned int rbc = 0u;

    for (int jt = 0; jt < JT_PER_CHUNK; ++jt) {
        const int j0 = j0base + jt * 16;
        float4 tj = T4[j0 + r];                       // column box (b128 load)
        v2f bfrag;
        bfrag.x = kh ? tj.z : tj.x;
        bfrag.y = kh ? tj.w : tj.y;
        float qj    = tj.x * tj.x + tj.y * tj.y + tj.z * tj.z + tj.w * tj.w;
        float tarea = (tj.z - tj.x + 1.f) * (tj.w - tj.y + 1.f);

        v8f c = {};
        // Gram tile G[m][n] = t_{i0+m} . t_{j0+n}  (16x16x4 f32 WMMA)
        v8f G = __builtin_amdgcn_wmma_f32_16x16x4_f32(
            false, afrag, false, bfrag, (short)0, c, false, false);

        // Same-target prefilter: dist2[m][n] = q_i + q_j - 2*G. One branch
        // per tile on min_v(dist2); bit-exact compare is the authority.
        float d2[8], dmin;
#pragma unroll
        for (int v = 0; v < 8; ++v) {
            d2[v] = qi[v] + qj - 2.f * G[v];
            dmin  = v ? fminf(dmin, d2[v]) : d2[v];
        }
        unsigned int samebits = 0u;
        if (dmin <= 64.f) {
#pragma unroll
            for (int v = 0; v < 8; ++v)
                samebits |= box_bits_eq(ta[v], tj) ? (1u << v) : 0u;
        }

        const int gj = j0 + r;
#pragma unroll
        for (int v = 0; v < 8; ++v) {
            const int gi = i0 + v + 8 * kh;
            // IoU with +1 pixel convention; v_rcp_f32 instead of IEEE div.
            float iw = fminf(p[v].z, tj.z) - fmaxf(p[v].x, tj.x) + 1.f;
            float ih = fminf(p[v].w, tj.w) - fmaxf(p[v].y, tj.y) + 1.f;
            iw = fmaxf(iw, 0.f);
            ih = fmaxf(ih, 0.f);
            float inter = iw * ih;
            float ov = inter * __builtin_amdgcn_rcpf(parea[v] + tarea - inter);

            bool same = (samebits >> v) & 1u;

            // RepGT: running (max ov, first argmax) packed as u64
            float ovg = same ? 0.f : ov;
            unsigned long long pk =
                ((unsigned long long)__float_as_uint(ovg) << 32) |
                (unsigned int)(~(unsigned int)gj);
            rmax[v] = (pk > rmax[v]) ? pk : rmax[v];

            // RepBox: strict lower triangle, different target, ov > 0.
            // smooth_ln(x, sigma=0) == x for x > 0, so just sum ov.
            bool hit = (gi > gj) & (!same) & (ov > 0.f);
            rbs += hit ? ov : 0.f;
            rbc += hit ? 1u : 0u;
        }
    }

    // Row max/argmax: reduce across the 16 lanes of each half-wave
    // (lanes 0-15 own rows v, lanes 16-31 own rows v+8), combine chunks
    // with a packed 64-bit atomicMax.
#pragma unroll
    for (int v = 0; v < 8; ++v) {
        unsigned long long x = rmax[v];
        for (int m = 8; m >= 1; m >>= 1) {
            unsigned long long y = shflxor_u64(x, m);
            x = (y > x) ? y : x;
        }
        if (r == 0)
            atomicMax(&rowmax[i0 + v + 8 * kh], x);
    }

    // RepBox partials: full-wave reduce, one atomic per wave.
    for (int m = 16; m >= 1; m >>= 1) {
        rbs += __shfl_xor(rbs, m, 32);
        rbc += __shfl_xor(rbc, m, 32);
    }
    if (lane == 0 && rbc) {
        atomicAdd(rb_sum, rbs);
        atomicAdd(rb_cnt, rbc);
    }
}

// ---------------- RepGT finalize: smooth_ln(IoG, 0.9) per row --------------
__global__ void repgt_finalize_kernel(const float* __restrict__ pred,
                                      const float* __restrict__ target,
                                      const unsigned long long* __restrict__ rowmax,
                                      float* __restrict__ rg_sum,
                                      unsigned int* __restrict__ rg_cnt) {
    int i = blockIdx.x * blockDim.x + threadIdx.x;
    float s = 0.f;
    unsigned int cnt = 0u;
    if (i < N_BOX) {
        unsigned long long pk = rowmax[i];
        unsigned int ovb = (unsigned int)(pk >> 32);
        if (ovb != 0u) {                       // max_ov > 0
            int arg = (int)(~(unsigned int)pk);
            const float4* P4 = (const float4*)pred;
            const float4* T4 = (const float4*)target;
            float4 pp = P4[i], g = T4[arg];
            // IoG: no +1 convention
            float iw = fmaxf(fminf(pp.z, g.z) - fmaxf(pp.x, g.x), 0.f);
            float ih = fmaxf(fminf(pp.w, g.w) - fmaxf(pp.y, g.y), 0.f);
            float garea = (g.z - g.x) * (g.w - g.y);
            float iog = iw * ih / garea;
            const float SIG = 0.9f;
            float sln;
            if (iog > SIG) {
                sln = (iog - SIG) / (1.f - SIG) - log1pf(-SIG);
            } else {
                float xc = fminf(fmaxf(iog, 0.f), 1.f - 1e-6f);
                sln = -log1pf(-xc);
            }
            s = sln;
            cnt = 1u;
        }
    }
    for (int m = 16; m >= 1; m >>= 1) {
        s   += __shfl_xor(s, m, 32);
        cnt += __shfl_xor(cnt, m, 32);
    }
    if ((threadIdx.x & 31) == 0 && cnt) {
        atomicAdd(rg_sum, s);
        atomicAdd(rg_cnt, cnt);
    }
}

// ---------------- broadcast final scalar -----------------------------------
__global__ void broadcast_kernel(float* __restrict__ out,
                                 const float* __restrict__ hdrf) {
    const unsigned int* u = (const unsigned int*)hdrf;
    float sl1 = hdrf[0] * (1.f / 16384.f);
    float rb  = u[2] ? hdrf[1] / (float)u[2] : 0.f;
    float rg  = u[4] ? hdrf[3] / (float)u[4] : 0.f;
    float scalar = sl1 + rg + rb;
    int i = blockIdx.x * blockDim.x + threadIdx.x;
    if (i < N_BOX) out[i] = scalar;
}

// ---------------------------------------------------------------------------
extern "C" void kernel_launch(void* const* d_in, const int* in_sizes, int n_in,
                              void* d_out, int out_size, void* d_ws, size_t ws_size,
                              hipStream_t stream) {
    (void)in_sizes; (void)n_in; (void)out_size; (void)ws_size;
    const float* pred   = (const float*)d_in[0];
    const float* target = (const float*)d_in[1];
    float* out = (float*)d_out;

    // workspace: [0 .. 32KB) rowmax u64[4096]; then 8 dwords of accumulators
    unsigned long long* rowmax = (unsigned long long*)d_ws;
    float*        hdrf = (float*)((char*)d_ws + N_BOX * sizeof(unsigned long long));
    unsigned int* hdru = (unsigned int*)hdrf;

    init_ws_kernel<<<16, 256, 0, stream>>>(rowmax, hdru);
    sl1_kernel<<<16, 256, 0, stream>>>(pred, target, hdrf + 0);

    dim3 grid(ITILES / WAVES_PER_BLK, JCHUNKS);   // (32, 8)
    pairwise_kernel<<<grid, 32 * WAVES_PER_BLK, 0, stream>>>(
        pred, target, rowmax, hdrf + 1, hdru + 2);

    repgt_finalize_kernel<<<16, 256, 0, stream>>>(pred, target, rowmax,
                                                  hdrf + 3, hdru + 4);
    broadcast_kernel<<<16, 256, 0, stream>>>(out, hdrf);
}